// My_loss_77008763617825
// MI455X (gfx1250) — compile-verified
//
#include <hip/hip_runtime.h>
#include <math.h>

// ---------------------------------------------------------------------------
// loss = mean_i softplus(-input[i, label[i]])
//   N = 131072 rows, C = 1000 classes, f32 input, i32 labels, scalar f32 out.
//
// Pass 1: one thread per row. Non-temporal gather of input[i*C+label[i]],
//         stable softplus, wave32 shuffle reduction + LDS across the 8 waves
//         of a 256-thread block, one f32 partial per block -> d_ws.
// Pass 2: single wave32. Lane-strided sum of the 512 partials (16 each),
//         then V_WMMA_F32_16X16X4_F32 with an all-ones A matrix:
//         D[m,n] = sum_k B[k,n]  =>  sum_n D[0,n] = sum of ALL elements of B,
//         independent of B's exact lane/VGPR striping. Final 16-lane shuffle
//         tree, scale by 1/N, store.
// ---------------------------------------------------------------------------

typedef __attribute__((ext_vector_type(2))) float v2f;
typedef __attribute__((ext_vector_type(8))) float v8f;

__device__ __forceinline__ float softplus_neg(float x) {
    // softplus(-x) = max(-x,0) + log1p(exp(-|x|))   (numerically stable)
    float t = -x;
    return fmaxf(t, 0.0f) + log1pf(expf(-fabsf(t)));
}

__global__ void gather_softplus_partial(const float* __restrict__ input,
                                        const int*   __restrict__ label,
                                        float*       __restrict__ partial,
                                        int N, int C) {
    const int i = blockIdx.x * blockDim.x + threadIdx.x;

    float v = 0.0f;
    if (i < N) {
        // Streamed once: non-temporal so we don't pollute WGP$/L2.
        int lab = __builtin_nontemporal_load(label + i);
        const float* p = input + (long long)i * (long long)C + lab;
        float x = __builtin_nontemporal_load(p);
        v = softplus_neg(x);
    }

    // wave32 tree reduction
    #pragma unroll
    for (int off = 16; off > 0; off >>= 1)
        v += __shfl_down(v, off, 32);

    __shared__ float wsum[8];          // 256 threads = 8 wave32
    const int lane = threadIdx.x & 31;
    const int wave = threadIdx.x >> 5;
    if (lane == 0) wsum[wave] = v;
    __syncthreads();

    if (threadIdx.x == 0) {
        float s = 0.0f;
        #pragma unroll
        for (int w = 0; w < 8; ++w) s += wsum[w];
        partial[blockIdx.x] = s;       // deterministic per-block partial
    }
}

// One wave32 (32 threads, EXEC all ones at the WMMA — no divergence before it).
__global__ void reduce_partials_wmma(const float* __restrict__ partial,
                                     float*       __restrict__ out,
                                     int NB, float invN) {
    const int lane = threadIdx.x;      // 0..31

    float s = 0.0f;
    for (int j = lane; j < NB; j += 32)   // NB=512 -> 16 adds/lane, fixed order
        s += partial[j];

    // A := ones(16x4): every A slot in every lane = 1.0  => layout-independent.
    v2f a; a.x = 1.0f; a.y = 1.0f;
    // B holds the 32 lane-partials somewhere in its 64 slots; rest are zero.
    // Total = sum over all B slots, which WMMA column-sums for us.
    v2f b; b.x = s; b.y = 0.0f;
    v8f c = {};
    // D = A x B + 0 ; emits v_wmma_f32_16x16x4_f32
    c = __builtin_amdgcn_wmma_f32_16x16x4_f32(
            /*neg_a=*/false, a, /*neg_b=*/false, b,
            /*c_mod=*/(short)0, c, /*reuse_a=*/false, /*reuse_b=*/false);

    // C/D layout: VGPR0 = D[0, n] for lanes 0..15 (lanes 16..31 hold D[8, n],
    // which with all-ones A is the identical column-sum row — ignored).
    float d0 = c[0];
    #pragma unroll
    for (int off = 8; off > 0; off >>= 1)
        d0 += __shfl_down(d0, off, 32);

    if (lane == 0) out[0] = d0 * invN;
}

extern "C" void kernel_launch(void* const* d_in, const int* in_sizes, int n_in,
                              void* d_out, int out_size, void* d_ws, size_t ws_size,
                              hipStream_t stream) {
    const float* input = (const float*)d_in[0];
    const int*   label = (const int*)d_in[1];
    // d_in[2] is class_num on device; we derive C host-side instead:
    const int N = in_sizes[1];                 // 131072
    const int C = in_sizes[0] / in_sizes[1];   // 1000

    float* out     = (float*)d_out;
    float* partial = (float*)d_ws;             // NB * 4 bytes

    const int threads = 256;                   // 8 wave32 per block
    const int NB = (N + threads - 1) / threads;

    gather_softplus_partial<<<NB, threads, 0, stream>>>(input, label, partial, N, C);
    reduce_partials_wmma<<<1, 32, 0, stream>>>(partial, out, NB, 1.0f / (float)N);
}